// CollectAttention_56753697849601
// MI455X (gfx1250) — compile-verified
//
#include <hip/hip_runtime.h>
#include <stdint.h>

// PSANet COLLECT gather, MI455X (gfx1250, wave32), TDM-staged.
// out[(oh*60+ow)*3600 + h*60 + w] = x[((oh+59-h)*119 + (ow+59-w))*3600 + h*60 + w]
//
// Block b = oh*60 + h (3600 blocks, 256 threads = 8 wave32).
// Phase 1: ONE Tensor Data Mover descriptor per block DMAs the 2D tile
//          (119 rows x 60 f32, row stride 3600 elems) into contiguous LDS.
//          Tracked by TENSORcnt; EXEC ignored; issued by wave 0 only.
// Phase 2: s_wait_tensorcnt 0 + workgroup barrier, then coalesced b32 stores
//          reading the LDS anti-diagonal (bank step 5 mod 64 -> conflict-free).

#define HDIM  60
#define WDIM  60
#define CW    119            // 2*W - 1
#define HW    3600           // H*W
#define NROWS 119            // dw rows staged per block

typedef uint32_t u32x4 __attribute__((ext_vector_type(4)));
typedef uint32_t u32x8 __attribute__((ext_vector_type(8)));

__global__ __launch_bounds__(256)
void psa_collect_kernel(const float* __restrict__ x, float* __restrict__ out) {
    const int b   = blockIdx.x;            // 0..3599
    const int oh  = b / HDIM;
    const int h   = b - oh * HDIM;
    const int dh  = oh + (HDIM - 1) - h;   // 0..118, fixed per block
    const int tid = threadIdx.x;

    __shared__ __align__(16) float smem[NROWS * WDIM];  // 28,560 B

    // ---- Phase 1: single TDM tile load (global -> LDS), wave 0 only ----
    if (tid < 32) {
        // LDS byte address: generic __shared__ pointer low 32 bits == LDS offset.
        const uint32_t lds_addr = (uint32_t)(uintptr_t)(&smem[0]);
        // Global byte address of tile start: element (dh*119 + 0)*3600 + h*60.
        const uint64_t gaddr =
            (uint64_t)(uintptr_t)(x + (size_t)dh * CW * HW + (size_t)h * WDIM);

        // D# group 0 (CDNA5 ISA): count=1, lds_addr,
        // global_addr[56:0], type=2 ("image") in bits [127:126].
        u32x4 g0;
        g0[0] = 1u;                                    // count=1, user D#
        g0[1] = lds_addr;                              // lds_addr[31:0]
        g0[2] = (uint32_t)gaddr;                       // global_addr[31:0]
        g0[3] = ((uint32_t)(gaddr >> 32) & 0x01FFFFFFu)
                | (2u << 30);                          // addr[56:32] | type=2

        // D# group 1 (§8.4): data_size=2 (4B), tensor_dim0=60, tensor_dim1=119,
        // tile_dim0=60, tile_dim1=119 (tile_dim2=0 -> 2D),
        // tensor_dim0_stride=3600 elems. No pad/multicast/iterate/barrier.
        u32x8 g1;
        g1[0] = 2u << 16;                              // data_size=4B
        g1[1] = (uint32_t)WDIM << 16;                  // tensor_dim0[15:0]
        g1[2] = (uint32_t)NROWS << 16;                 // tensor_dim1[15:0]
        g1[3] = (uint32_t)WDIM << 16;                  // tile_dim0
        g1[4] = (uint32_t)NROWS;                       // tile_dim1 (tile_dim2=0)
        g1[5] = (uint32_t)HW;                          // tensor_dim0_stride[31:0]
        g1[6] = 0u;                                    // stride[47:32], dim1_stride
        g1[7] = 0u;

        asm volatile("tensor_load_to_lds %0, %1"
                     :
                     : "s"(g0), "s"(g1)
                     : "memory");
        // Issuing wave drains TENSORcnt before signaling the barrier.
        asm volatile("s_wait_tensorcnt 0" ::: "memory");
    }
    __syncthreads();

    // ---- Phase 2: coalesced stores from the LDS anti-diagonal ----
    float* outb = out + (size_t)(oh * WDIM) * HW + (size_t)h * WDIM;
    for (int i = tid; i < HW; i += 256) {
        const int ow = i / WDIM;
        const int w  = i - ow * WDIM;
        const int dw = ow + (WDIM - 1) - w;            // 0..118
        outb[(size_t)ow * HW + w] = smem[dw * WDIM + w];
    }
}

extern "C" void kernel_launch(void* const* d_in, const int* in_sizes, int n_in,
                              void* d_out, int out_size, void* d_ws, size_t ws_size,
                              hipStream_t stream) {
    (void)in_sizes; (void)n_in; (void)d_ws; (void)ws_size; (void)out_size;
    const float* x = (const float*)d_in[0];
    float* out    = (float*)d_out;
    psa_collect_kernel<<<dim3(HW), dim3(256), 0, stream>>>(x, out);
}